// FactorGraphMsgPassingLayer_NoDoubleCounting_38113539784904
// MI455X (gfx1250) — compile-verified
//
#include <hip/hip_runtime.h>
#include <cstdint>

// ---------------------------------------------------------------------------
// FactorGraph BP step, C=2. Memory-bound (~300MB traffic, ~0.25 GFLOP):
// design is 3 fused kernels, vectorized b64/b128 access, f32 atomics only for
// the var-belief segment-sum (random var_idx), no atomics for the factor
// segment-sum (factor f owns edges 2f,2f+1 by construction).
// CDNA5 path: TDM (tensor_load_to_lds) stages MLP weights into LDS, waited
// with s_wait_tensorcnt; weights are then read via DS (LDS) loads.
// ---------------------------------------------------------------------------

#define SHIFT_EPS 1e-19f

typedef uint32_t u32;
typedef __attribute__((ext_vector_type(4))) unsigned int v4u;
typedef __attribute__((ext_vector_type(8))) int v8i;
typedef __attribute__((ext_vector_type(4))) int v4i;

__device__ __forceinline__ float lse2(float a, float b) {
    // logsumexp of 2 elements, max-trick + log1p (matches jax lse numerics)
    float mx = fmaxf(a, b);
    float mn = fminf(a, b);
    return mx + log1pf(expf(mn - mx));
}

// TDM 1-D copy of `ndw` dwords from global to LDS byte-offset `lds_off`.
// D# per CDNA5 ISA ch.8: group0 = {count=1, lds_addr, global_addr, type=2},
// group1 = {data_size=4B, tensor_dim0=ndw, tile_dim0=ndw, dim0_stride=ndw}.
__device__ __forceinline__ void tdm_load_dwords(const void* gptr, u32 lds_off, u32 ndw) {
    uintptr_t ga = (uintptr_t)gptr;
    v4u g0;
    g0.x = 1u;                                            // count=1, user desc
    g0.y = lds_off;                                       // lds_addr (bytes)
    g0.z = (u32)ga;                                       // global_addr[31:0]
    g0.w = ((u32)(ga >> 32) & 0x01FFFFFFu) | (2u << 30);  // addr[56:32] | type=2
    v8i g1;
    g1[0] = 0x20000;                     // data_size=2 (4 bytes), no mask/flags
    g1[1] = (int)((ndw & 0xFFFFu) << 16);// tensor_dim0 low16 @ bits[63:48]
    g1[2] = 0;                           // tensor_dim0 hi16, tensor_dim1 lo16
    g1[3] = (int)((ndw & 0xFFFFu) << 16);// tile_dim0 @ bits[127:112]
    g1[4] = 0;                           // tile_dim1=0 (unused), tile_dim2=0
    g1[5] = (int)ndw;                    // tensor_dim0_stride low32
    g1[6] = 0;
    g1[7] = 0;
    v4i z4 = {0, 0, 0, 0};
    v8i z8 = {0, 0, 0, 0, 0, 0, 0, 0};
    // 6-arg form on this toolchain: (g0, g1, g2, g3, <zero-filled v8i>, cpol)
    __builtin_amdgcn_tensor_load_to_lds(g0, g1, z4, z4, z8, 0);
}

// Stage W(4) b(2) W2(4) b2(2) = 12 floats into LDS via TDM; all waves wait.
__device__ __forceinline__ void stage_weights(float* wlds,
                                              const float* W1, const float* B1,
                                              const float* W2, const float* B2) {
    if (threadIdx.x == 0) {
        u32 base = (u32)(uintptr_t)&wlds[0];   // low 32 bits of flat LDS addr == LDS offset
        tdm_load_dwords(W1, base + 0u, 4);
        tdm_load_dwords(B1, base + 16u, 2);
        tdm_load_dwords(W2, base + 24u, 4);
        tdm_load_dwords(B2, base + 40u, 2);
    }
    __builtin_amdgcn_s_wait_tensorcnt(0);      // TDM ignores EXEC; every wave drains
    __syncthreads();
}

// exp-space MLP: log( relu( relu(exp(x) @ W1^T + b1) @ W2^T + b2 ) + SHIFT )
__device__ __forceinline__ void mlp2(const float* w, float x0, float x1,
                                     float& r0, float& r1) {
    float m0 = expf(x0), m1 = expf(x1);
    float h0 = fmaxf(0.f, fmaf(m0, w[0], fmaf(m1, w[1], w[4])));
    float h1 = fmaxf(0.f, fmaf(m0, w[2], fmaf(m1, w[3], w[5])));
    float o0 = fmaxf(0.f, fmaf(h0, w[6], fmaf(h1, w[7], w[10]))) + SHIFT_EPS;
    float o1 = fmaxf(0.f, fmaf(h0, w[8], fmaf(h1, w[9], w[11]))) + SHIFT_EPS;
    r0 = logf(o0);
    r1 = logf(o1);
}

__global__ void __launch_bounds__(256) zero_kernel(float* __restrict__ p, int n) {
    int i = blockIdx.x * blockDim.x + threadIdx.x;
    if (i < n) p[i] = 0.f;
}

// Kernel 1: per-edge factor->var messages + atomic segment-sum into var beliefs.
__global__ void __launch_bounds__(256)
ftv_kernel(const float* __restrict__ prv_vtf,   // [E,2]
           const float* __restrict__ prv_ftv,   // [E,2]
           const float* __restrict__ prv_fb,    // [F,2,2]
           const float* __restrict__ W5, const float* __restrict__ b5,
           const float* __restrict__ W6, const float* __restrict__ b6,
           const float* __restrict__ alpha3_p,
           const int* __restrict__ fac_idx,
           const int* __restrict__ var_idx,
           const int* __restrict__ dim_arr,
           float* __restrict__ out_ftv,         // [E,2]
           float* __restrict__ vb_accum,        // [V,2] (pre-zeroed)
           int E) {
    __shared__ float wlds[12];
    stage_weights(wlds, W5, b5, W6, b6);

    int e = blockIdx.x * blockDim.x + threadIdx.x;
    if (e >= E) return;
    const float a3 = alpha3_p[0];

    int f   = fac_idx[e];
    int dim = dim_arr[e];
    float4 fb  = ((const float4*)prv_fb)[f];     // [f][0][0],[0][1],[1][0],[1][1]
    float2 msg = ((const float2*)prv_vtf)[e];

    // marginalize the dim the edge does NOT own, in log space:
    // dim==0: marg[a] = lse_b(fb[a][b]) - msg[a]; dim==1: marg[b] = lse_a - msg[b]
    float marg0, marg1;
    if (dim == 0) {
        marg0 = lse2(fb.x, fb.y) - msg.x;
        marg1 = lse2(fb.z, fb.w) - msg.y;
    } else {
        marg0 = lse2(fb.x, fb.z) - msg.x;
        marg1 = lse2(fb.y, fb.w) - msg.y;
    }

    float r0, r1;
    mlp2(wlds, marg0, marg1, r0, r1);
    float p0 = a3 * marg0 + (1.f - a3) * r0;
    float p1 = a3 * marg1 + (1.f - a3) * r1;

    float2 pf = ((const float2*)prv_ftv)[e];
    float d0 = 0.5f * p0 + 0.5f * pf.x;          // A_FTOV damping
    float d1 = 0.5f * p1 + 0.5f * pf.y;
    float nrm = lse2(d0, d1);
    float t0 = d0 - nrm, t1 = d1 - nrm;

    ((float2*)out_ftv)[e] = make_float2(t0, t1);

    int v = var_idx[e];
    atomicAdd(&vb_accum[2 * v + 0], t0);
    atomicAdd(&vb_accum[2 * v + 1], t1);
}

// Kernel 2: per-FACTOR var->factor messages for both owned edges (2f, 2f+1)
// plus factor-belief assembly (no atomics; fac_idx = repeat(arange(F),2)).
__global__ void __launch_bounds__(256)
vtf_kernel(const float* __restrict__ prv_vtf,   // [E,2]
           const float* __restrict__ ftv,       // [E,2] (from kernel 1)
           const float* __restrict__ vb,        // [V,2]
           const float* __restrict__ pot,       // [F,2,2]
           const float* __restrict__ W7, const float* __restrict__ b7,
           const float* __restrict__ W8, const float* __restrict__ b8,
           const float* __restrict__ alpha4_p,
           const int* __restrict__ var_idx,
           const int* __restrict__ dim_arr,
           float* __restrict__ out_vtf,         // [E,2]
           float* __restrict__ out_fb,          // [F,2,2]
           int F) {
    __shared__ float wlds[12];
    stage_weights(wlds, W7, b7, W8, b8);

    int f = blockIdx.x * blockDim.x + threadIdx.x;
    if (f >= F) return;
    const float a4 = alpha4_p[0];

    float4 ftv2 = ((const float4*)ftv)[f];       // edges 2f (x,y) and 2f+1 (z,w)
    float4 pv2  = ((const float4*)prv_vtf)[f];
    int va = var_idx[2 * f + 0];
    int vb_i = var_idx[2 * f + 1];
    int dA = dim_arr[2 * f + 0];
    int dB = dim_arr[2 * f + 1];
    float2 vba = ((const float2*)vb)[va];        // gathered var beliefs
    float2 vbb = ((const float2*)vb)[vb_i];

    // edge A = 2f
    float exA0 = vba.x - ftv2.x, exA1 = vba.y - ftv2.y;
    float rA0, rA1;
    mlp2(wlds, exA0, exA1, rA0, rA1);
    float pA0 = a4 * exA0 + (1.f - a4) * rA0;
    float pA1 = a4 * exA1 + (1.f - a4) * rA1;
    float dA0 = 0.5f * pA0 + 0.5f * pv2.x;       // A_VTOF damping
    float dA1 = 0.5f * pA1 + 0.5f * pv2.y;
    float nA = lse2(dA0, dA1);
    float A0 = dA0 - nA, A1 = dA1 - nA;

    // edge B = 2f+1
    float exB0 = vbb.x - ftv2.z, exB1 = vbb.y - ftv2.w;
    float rB0, rB1;
    mlp2(wlds, exB0, exB1, rB0, rB1);
    float pB0 = a4 * exB0 + (1.f - a4) * rB0;
    float pB1 = a4 * exB1 + (1.f - a4) * rB1;
    float dB0 = 0.5f * pB0 + 0.5f * pv2.z;
    float dB1 = 0.5f * pB1 + 0.5f * pv2.w;
    float nB = lse2(dB0, dB1);
    float B0 = dB0 - nB, B1 = dB1 - nB;

    float4 ov; ov.x = A0; ov.y = A1; ov.z = B0; ov.w = B1;
    ((float4*)out_vtf)[f] = ov;

    // expand(msg, dim)[a][b] = (dim==0 ? msg[a] : msg[b]); sum both edges + pot
    float cA01 = (dA == 0) ? A0 : A1;
    float cA10 = (dA == 0) ? A1 : A0;
    float cB01 = (dB == 0) ? B0 : B1;
    float cB10 = (dB == 0) ? B1 : B0;
    float4 p4 = ((const float4*)pot)[f];
    float4 ofb;
    ofb.x = p4.x + A0 + B0;
    ofb.y = p4.y + cA01 + cB01;
    ofb.z = p4.z + cA10 + cB10;
    ofb.w = p4.w + A1 + B1;
    ((float4*)out_fb)[f] = ofb;
}

extern "C" void kernel_launch(void* const* d_in, const int* in_sizes, int n_in,
                              void* d_out, int out_size, void* d_ws, size_t ws_size,
                              hipStream_t stream) {
    // setup_inputs() order
    const float* prv_vtf = (const float*)d_in[0];
    const float* prv_ftv = (const float*)d_in[1];
    const float* prv_fb  = (const float*)d_in[2];
    const float* pot     = (const float*)d_in[3];
    const float* W5 = (const float*)d_in[4];
    const float* b5 = (const float*)d_in[5];
    const float* W6 = (const float*)d_in[6];
    const float* b6 = (const float*)d_in[7];
    const float* W7 = (const float*)d_in[8];
    const float* b7 = (const float*)d_in[9];
    const float* W8 = (const float*)d_in[10];
    const float* b8 = (const float*)d_in[11];
    const float* a3 = (const float*)d_in[12];
    const float* a4 = (const float*)d_in[13];
    const int* fac_idx = (const int*)d_in[14];
    const int* var_idx = (const int*)d_in[15];
    const int* dims    = (const int*)d_in[16];

    const int E2  = in_sizes[0];           // E*C
    const int E   = E2 / 2;
    const int FCC = in_sizes[2];           // F*C*C
    const int F   = FCC / 4;
    const int VC  = out_size - 2 * E2 - FCC;  // V*C

    // outputs concatenated in return order: vtf, ftv, var_beliefs, factor_beliefs
    float* out_vtf = (float*)d_out;
    float* out_ftv = out_vtf + E2;
    float* out_vb  = out_ftv + E2;
    float* out_fb  = out_vb + VC;

    const int B = 256;  // 8 wave32s per block
    zero_kernel<<<(VC + B - 1) / B, B, 0, stream>>>(out_vb, VC);
    ftv_kernel<<<(E + B - 1) / B, B, 0, stream>>>(
        prv_vtf, prv_ftv, prv_fb, W5, b5, W6, b6, a3,
        fac_idx, var_idx, dims, out_ftv, out_vb, E);
    vtf_kernel<<<(F + B - 1) / B, B, 0, stream>>>(
        prv_vtf, out_ftv, out_vb, pot, W7, b7, W8, b8, a4,
        var_idx, dims, out_vtf, out_fb, F);
}